// SE3GNNPredictor_29884382446300
// MI455X (gfx1250) — compile-verified
//
#include <hip/hip_runtime.h>
#include <hip/hip_bf16.h>

#define HID 32
#define SE3N 0.17677669529663687f /* 1/sqrt(32) */

typedef float v2f __attribute__((ext_vector_type(2)));
typedef float v8f __attribute__((ext_vector_type(8)));

// D(16x16,f32) = A(16x4,f32) x B(4x16,f32) + C   — CDNA5 V_WMMA_F32_16X16X4_F32
__device__ __forceinline__ v8f wmma4(v2f a, v2f b, v8f c) {
  return __builtin_amdgcn_wmma_f32_16x16x4_f32(false, a, false, b, (short)0, c,
                                               false, false);
}

__device__ __forceinline__ float leaky(float x) { return x > 0.f ? x : 0.01f * x; }

// ---------------- zero the accumulators (agg1: 2n, agg2: 32n contiguous) ----
__global__ void k_init(float* __restrict__ p, int tot) {
  int i = blockIdx.x * blockDim.x + threadIdx.x;
  if (i < tot) p[i] = 0.f;
}

// ---------------- edge pass 1: agg1[dst] += pos[src, :2] --------------------
__global__ void k_edge1(const int* __restrict__ ei, const float* __restrict__ pos,
                        float* __restrict__ agg1, int nE) {
  int e = blockIdx.x * blockDim.x + threadIdx.x;
  if (e >= nE) return;
  int s = ei[e];
  int d = ei[nE + e];
  atomicAdd(&agg1[d * 2 + 0], pos[s * 3 + 0]);
  atomicAdd(&agg1[d * 2 + 1], pos[s * 3 + 1]);
}

// ---------------- node pass 1: h1 = leaky(agg1@Wl1+bl1+x@Wr1) @ Wse1 * NORM -
__global__ void k_node1(const float* __restrict__ pos, const float* __restrict__ agg1,
                        const float* __restrict__ Wl1, const float* __restrict__ bl1,
                        const float* __restrict__ Wr1, const float* __restrict__ Wse1,
                        float* __restrict__ h1, int n, int numTiles) {
  int wave = blockIdx.x * (blockDim.x >> 5) + (threadIdx.x >> 5);
  int lane = threadIdx.x & 31;
  if (wave >= numTiles) return;              // wave-uniform exit, EXEC stays all-1
  int base = wave * 16;
  int m  = lane & 15;
  int kh = lane >> 4;
  int nn = lane & 15;
  int row = base + m; if (row > n - 1) row = n - 1;

  float a0 = agg1[row * 2 + 0], a1 = agg1[row * 2 + 1];
  float x0 = pos[row * 3 + 0],  x1 = pos[row * 3 + 1];

  // A fragments: lane holds rows m, K pair {4kk+2kh, 4kk+2kh+1}
  v2f A[8];
#pragma unroll
  for (int kk = 0; kk < 8; ++kk) {
#pragma unroll
    for (int j = 0; j < 2; ++j) {
      int f = 4 * kk + 2 * kh + j;
      float t = a0 * Wl1[f] + a1 * Wl1[HID + f] + bl1[f]
              + x0 * Wr1[f] + x1 * Wr1[HID + f];
      A[kk][j] = leaky(t);
    }
  }
  v8f C0 = {}, C1 = {};
#pragma unroll
  for (int kk = 0; kk < 8; ++kk) {
    int k0 = 4 * kk + 2 * kh;
    v2f B0; B0.x = Wse1[k0 * HID + nn];      B0.y = Wse1[(k0 + 1) * HID + nn];
    v2f B1; B1.x = Wse1[k0 * HID + 16 + nn]; B1.y = Wse1[(k0 + 1) * HID + 16 + nn];
    C0 = wmma4(A[kk], B0, C0);
    C1 = wmma4(A[kk], B1, C1);
  }
  // D layout: vgpr v -> row v+8*kh, col nn (+16 for tile 1)
#pragma unroll
  for (int v = 0; v < 8; ++v) {
    int r = base + v + 8 * kh;
    if (r < n) {
      h1[(size_t)r * HID + nn]      = C0[v] * SE3N;
      h1[(size_t)r * HID + 16 + nn] = C1[v] * SE3N;
    }
  }
}

// ---------------- edge pass 2: agg2[dst] += h1[src]  (8 lanes per edge) -----
__global__ void k_edge2(const int* __restrict__ ei, const float* __restrict__ h1,
                        float* __restrict__ agg2, int nE) {
  long long tid = (long long)blockIdx.x * blockDim.x + threadIdx.x;
  int e = (int)(tid >> 3);
  int g = (int)(tid & 7);
  if (e >= nE) return;
  int s = ei[e];
  int d = ei[nE + e];
  const float4 v = ((const float4*)(h1 + (size_t)s * HID))[g];
  float* ap = agg2 + (size_t)d * HID + g * 4;
  atomicAdd(ap + 0, v.x);
  atomicAdd(ap + 1, v.y);
  atomicAdd(ap + 2, v.z);
  atomicAdd(ap + 3, v.w);
}

// ---------------- node pass 2 + head ---------------------------------------
__global__ void k_node2(const float* __restrict__ h1, const float* __restrict__ agg2,
                        const float* __restrict__ Wl2, const float* __restrict__ bl2,
                        const float* __restrict__ Wr2, const float* __restrict__ Wse2,
                        const float* __restrict__ W3, const float* __restrict__ b3,
                        const float* __restrict__ W4, const float* __restrict__ b4,
                        const float* __restrict__ alphap,
                        float* __restrict__ out, int n, int numTiles) {
  __shared__ float tile[4][16][HID + 1];
  int wid  = threadIdx.x >> 5;
  int wave = blockIdx.x * (blockDim.x >> 5) + wid;
  int lane = threadIdx.x & 31;
  bool live = wave < numTiles;
  int tw = live ? wave : numTiles - 1;       // keep all waves alive for barriers
  int base = tw * 16;
  int m  = lane & 15;
  int kh = lane >> 4;
  int nn = lane & 15;
  int row = base + m; if (row > n - 1) row = n - 1;
  float alpha = *alphap;

  // ---- S = agg2 @ Wl2 + h1 @ Wr2  (accumulated in one C chain) ----
  v8f C0 = {}, C1 = {};
#pragma unroll
  for (int kk = 0; kk < 8; ++kk) {
    int k0 = 4 * kk + 2 * kh;
    v2f A; A.x = agg2[(size_t)row * HID + k0]; A.y = agg2[(size_t)row * HID + k0 + 1];
    v2f B0; B0.x = Wl2[k0 * HID + nn];      B0.y = Wl2[(k0 + 1) * HID + nn];
    v2f B1; B1.x = Wl2[k0 * HID + 16 + nn]; B1.y = Wl2[(k0 + 1) * HID + 16 + nn];
    C0 = wmma4(A, B0, C0);
    C1 = wmma4(A, B1, C1);
  }
#pragma unroll
  for (int kk = 0; kk < 8; ++kk) {
    int k0 = 4 * kk + 2 * kh;
    v2f A; A.x = h1[(size_t)row * HID + k0]; A.y = h1[(size_t)row * HID + k0 + 1];
    v2f B0; B0.x = Wr2[k0 * HID + nn];      B0.y = Wr2[(k0 + 1) * HID + nn];
    v2f B1; B1.x = Wr2[k0 * HID + 16 + nn]; B1.y = Wr2[(k0 + 1) * HID + 16 + nn];
    C0 = wmma4(A, B0, C0);
    C1 = wmma4(A, B1, C1);
  }
  // h2 = leaky(S + bl2)  — bias column fixed per lane
  float bb0 = bl2[nn], bb1 = bl2[16 + nn];
#pragma unroll
  for (int v = 0; v < 8; ++v) { C0[v] = leaky(C0[v] + bb0); C1[v] = leaky(C1[v] + bb1); }

  // ---- C-layout -> A-layout via LDS; h2s = h2 @ Wse2 ----
  __syncthreads();
#pragma unroll
  for (int v = 0; v < 8; ++v) {
    tile[wid][v + 8 * kh][nn]      = C0[v];
    tile[wid][v + 8 * kh][16 + nn] = C1[v];
  }
  __syncthreads();
  v2f A2[8];
#pragma unroll
  for (int kk = 0; kk < 8; ++kk) {
    int k0 = 4 * kk + 2 * kh;
    A2[kk].x = tile[wid][m][k0]; A2[kk].y = tile[wid][m][k0 + 1];
  }
  v8f D0 = {}, D1 = {};
#pragma unroll
  for (int kk = 0; kk < 8; ++kk) {
    int k0 = 4 * kk + 2 * kh;
    v2f B0; B0.x = Wse2[k0 * HID + nn];      B0.y = Wse2[(k0 + 1) * HID + nn];
    v2f B1; B1.x = Wse2[k0 * HID + 16 + nn]; B1.y = Wse2[(k0 + 1) * HID + 16 + nn];
    D0 = wmma4(A2[kk], B0, D0);
    D1 = wmma4(A2[kk], B1, D1);
  }
  // x_skip = alpha*h1 + h2s*NORM  (C layout)
#pragma unroll
  for (int v = 0; v < 8; ++v) {
    int r = base + v + 8 * kh; int rr = r > n - 1 ? n - 1 : r;
    D0[v] = alpha * h1[(size_t)rr * HID + nn]      + D0[v] * SE3N;
    D1[v] = alpha * h1[(size_t)rr * HID + 16 + nn] + D1[v] * SE3N;
  }

  // ---- again through LDS; x_out = relu(x_skip @ W3 + b3) ----
  __syncthreads();
#pragma unroll
  for (int v = 0; v < 8; ++v) {
    tile[wid][v + 8 * kh][nn]      = D0[v];
    tile[wid][v + 8 * kh][16 + nn] = D1[v];
  }
  __syncthreads();
#pragma unroll
  for (int kk = 0; kk < 8; ++kk) {
    int k0 = 4 * kk + 2 * kh;
    A2[kk].x = tile[wid][m][k0]; A2[kk].y = tile[wid][m][k0 + 1];
  }
  v8f E0 = {}, E1 = {};
#pragma unroll
  for (int kk = 0; kk < 8; ++kk) {
    int k0 = 4 * kk + 2 * kh;
    v2f B0; B0.x = W3[k0 * HID + nn];      B0.y = W3[(k0 + 1) * HID + nn];
    v2f B1; B1.x = W3[k0 * HID + 16 + nn]; B1.y = W3[(k0 + 1) * HID + 16 + nn];
    E0 = wmma4(A2[kk], B0, E0);
    E1 = wmma4(A2[kk], B1, E1);
  }
  float c30 = b3[nn], c31 = b3[16 + nn];
  __syncthreads();
#pragma unroll
  for (int v = 0; v < 8; ++v) {
    float e0 = E0[v] + c30; e0 = e0 > 0.f ? e0 : 0.f;
    float e1 = E1[v] + c31; e1 = e1 > 0.f ? e1 : 0.f;
    tile[wid][v + 8 * kh][nn]      = e0;
    tile[wid][v + 8 * kh][16 + nn] = e1;
  }
  __syncthreads();
  // pred = x_out @ W4 + b4  (GEMV, lanes 0..15)
  if (lane < 16 && live) {
    int r = base + lane;
    if (r < n) {
      float acc = b4[0];
#pragma unroll
      for (int k = 0; k < HID; ++k) acc += tile[wid][lane][k] * W4[k];
      out[r] = acc;
    }
  }
}

extern "C" void kernel_launch(void* const* d_in, const int* in_sizes, int n_in,
                              void* d_out, int out_size, void* d_ws, size_t ws_size,
                              hipStream_t stream) {
  const float* pos   = (const float*)d_in[0];
  const int*   ei    = (const int*)  d_in[1];
  const float* Wl1   = (const float*)d_in[2];
  const float* bl1   = (const float*)d_in[3];
  const float* Wr1   = (const float*)d_in[4];
  const float* Wl2   = (const float*)d_in[5];
  const float* bl2   = (const float*)d_in[6];
  const float* Wr2   = (const float*)d_in[7];
  const float* Wse1  = (const float*)d_in[8];
  const float* Wse2  = (const float*)d_in[9];
  const float* W3    = (const float*)d_in[10];
  const float* b3    = (const float*)d_in[11];
  const float* W4    = (const float*)d_in[12];
  const float* b4    = (const float*)d_in[13];
  const float* alphap= (const float*)d_in[14];

  int n  = in_sizes[0] / 3;
  int nE = in_sizes[1] / 2;
  int numTiles = (n + 15) / 16;

  float* agg1 = (float*)d_ws;                       // 2n  f32
  float* agg2 = agg1 + 2 * (size_t)n;               // 32n f32
  float* h1   = agg2 + (size_t)HID * n;             // 32n f32

  int tot = 34 * n;
  k_init<<<(tot + 255) / 256, 256, 0, stream>>>(agg1, tot);

  k_edge1<<<(nE + 255) / 256, 256, 0, stream>>>(ei, pos, agg1, nE);

  k_node1<<<(numTiles + 3) / 4, 128, 0, stream>>>(pos, agg1, Wl1, bl1, Wr1, Wse1,
                                                  h1, n, numTiles);

  long long work2 = (long long)nE * 8;
  k_edge2<<<(unsigned)((work2 + 255) / 256), 256, 0, stream>>>(ei, h1, agg2, nE);

  k_node2<<<(numTiles + 3) / 4, 128, 0, stream>>>(h1, agg2, Wl2, bl2, Wr2, Wse2,
                                                  W3, b3, W4, b4, alphap,
                                                  (float*)d_out, n, numTiles);
}